// Model_19679540150345
// MI455X (gfx1250) — compile-verified
//
#include <hip/hip_runtime.h>
#include <hip/hip_bf16.h>
#include <stdint.h>

// ---------------------------------------------------------------------------
// y = 0.75 * x @ colsum(weight)   with x:[8192,8192], weight:[8192,8192] f32
// Bandwidth-bound: 512 MB streamed @ 23.3 TB/s => ~22 us floor.
// Pass 1: partial column sums of weight (deterministic, no atomics)
// Pass 2: fold 32 partials -> wcs[8192]
// Pass 3: per-row dot(x_row, wcs); wcs staged to LDS via gfx1250 async copy,
//         final 16-wide reduction done with one v_wmma_f32_16x16x4_f32.
// ---------------------------------------------------------------------------

typedef __attribute__((ext_vector_type(2))) float v2f;
typedef __attribute__((ext_vector_type(4))) float v4f;
typedef __attribute__((ext_vector_type(8))) float v8f;

#define KDIM 8192
#define NDIM 8192
#define MDIM 8192
#define ROWSPLIT 32           // row chunks for pass-1 partials (scratch = 1 MB)

// ---- Pass 1: each block sums 256 rows over a 1024-column strip -------------
__global__ __launch_bounds__(256) void colsum_partial_k(const float* __restrict__ w,
                                                        float* __restrict__ part) {
  const int col  = blockIdx.x * 1024 + threadIdx.x * 4;
  const int row0 = blockIdx.y * (NDIM / ROWSPLIT);
  const float* p = w + (size_t)row0 * KDIM + col;
  v4f acc = {0.f, 0.f, 0.f, 0.f};
#pragma unroll 8
  for (int i = 0; i < NDIM / ROWSPLIT; ++i) {      // 256 coalesced b128 loads
    v4f v = *(const v4f*)p;
    acc += v;
    p += KDIM;
  }
  *(v4f*)(part + (size_t)blockIdx.y * KDIM + col) = acc;
}

// ---- Pass 2: reduce the 32 partial rows into wcs[8192] ---------------------
__global__ __launch_bounds__(256) void colsum_reduce_k(const float* __restrict__ part,
                                                       float* __restrict__ wcs) {
  const int col = (blockIdx.x * 256 + threadIdx.x) * 4;
  v4f acc = {0.f, 0.f, 0.f, 0.f};
#pragma unroll
  for (int i = 0; i < ROWSPLIT; ++i)
    acc += *(const v4f*)(part + (size_t)i * KDIM + col);
  *(v4f*)(wcs + col) = acc;
}

// ---- Pass 3: 16 rows per block; dot each row with LDS-resident wcs ---------
__global__ __launch_bounds__(256) void matvec_wmma_k(const float* __restrict__ x,
                                                     const float* __restrict__ wcs,
                                                     float* __restrict__ out) {
  __shared__ float wlds[KDIM];   // 32 KB staged copy of the colsum vector
  __shared__ float P[16][16];    // per-block partial-dot matrix

  // Stage wcs into LDS with CDNA5 async global->LDS copies (ASYNCcnt path).
  {
    unsigned lbase = (unsigned)(uintptr_t)&wlds[0];  // low 32 bits = LDS offset
#pragma unroll
    for (int i = 0; i < KDIM / (256 * 4); ++i) {     // 8 x 16B per thread
      const int f = (threadIdx.x + i * 256) * 4;     // float index
      uint64_t g  = (uint64_t)(wcs + f);
      unsigned l  = lbase + (unsigned)f * 4u;        // LDS byte address
      asm volatile("global_load_async_to_lds_b128 %0, %1, off"
                   :: "v"(l), "v"(g) : "memory");
    }
    asm volatile("s_wait_asynccnt 0x0" ::: "memory");
  }
  __syncthreads();

  // Thread (r,c): row r of this block's 16-row tile, column-chunk c.
  const int r   = threadIdx.x >> 4;
  const int c   = threadIdx.x & 15;
  const int row = blockIdx.x * 16 + r;
  const float* xr = x + (size_t)row * KDIM;

  v4f acc = {0.f, 0.f, 0.f, 0.f};
#pragma unroll 4
  for (int k = c * 4; k < KDIM; k += 64) {   // half-wave reads 256B contiguous
    v4f xv = *(const v4f*)(xr + k);
    v4f wv = *(const v4f*)(&wlds[k]);
    acc += xv * wv;
  }
  P[r][c] = acc.x + acc.y + acc.z + acc.w;
  __syncthreads();

  // Wave 0 finishes: fold 16 partials/row to 4, then D = ones_row_A x Q.
  // A is 16x4 (lane m=lane&15 holds A[m][2h],A[m][2h+1], h=lane>>4):
  //   A[0][k]=1, A[m>0][k]=0  ->  D[0][n] = sum_k Q[k][n] = y[row n].
  if (threadIdx.x < 32) {
    const int n = threadIdx.x & 15;
    const int h = threadIdx.x >> 4;
    v2f a, b;
    const float one = (n == 0) ? 1.0f : 0.0f;
    a.x = one; a.y = one;
    const int k0 = 2 * h;
    b.x = P[n][k0]     + P[n][k0 + 4] + P[n][k0 + 8] + P[n][k0 + 12];
    b.y = P[n][k0 + 1] + P[n][k0 + 5] + P[n][k0 + 9] + P[n][k0 + 13];
    v8f cacc = {};
    v8f d = __builtin_amdgcn_wmma_f32_16x16x4_f32(
        /*neg_a=*/false, a, /*neg_b=*/false, b,
        /*c_mod=*/(short)0, cacc, /*reuse_a=*/false, /*reuse_b=*/false);
    // D[0][n] lives in VGPR0 (component 0) of lanes 0..15.
    if (h == 0) out[blockIdx.x * 16 + n] = 0.75f * d[0];
  }
}

extern "C" void kernel_launch(void* const* d_in, const int* in_sizes, int n_in,
                              void* d_out, int out_size, void* d_ws, size_t ws_size,
                              hipStream_t stream) {
  const float* x = (const float*)d_in[0];      // [M,K]
  const float* w = (const float*)d_in[1];      // [N,K]
  float* out  = (float*)d_out;                 // [M]
  float* part = (float*)d_ws;                  // ROWSPLIT*K floats = 1 MB
  float* wcs  = part + (size_t)ROWSPLIT * KDIM; // K floats = 32 KB

  colsum_partial_k<<<dim3(KDIM / 1024, ROWSPLIT), 256, 0, stream>>>(w, part);
  colsum_reduce_k<<<KDIM / 1024, 256, 0, stream>>>(part, wcs);
  matvec_wmma_k<<<MDIM / 16, 256, 0, stream>>>(x, wcs, out);
}